// SphericalConvTranspose_16509854286075
// MI455X (gfx1250) — compile-verified
//
#include <hip/hip_runtime.h>
#include <math.h>

typedef __attribute__((ext_vector_type(2))) float v2f;
typedef __attribute__((ext_vector_type(8))) float v8f;

// Problem constants (from reference setup_inputs)
constexpr int BB   = 8;      // batch
constexpr int CIN  = 64;     // input channels
constexpr int NPTS = 16384;  // points
constexpr int KADJ = 17;     // adj last dim (divisor)
constexpr int COUT = 64;     // output channels
constexpr int DDIM = 65;     // frame dim (Cin + 1)
constexpr int DP   = 68;     // frame dim padded to multiple of 4 (WMMA K-steps = 17)
constexpr int PTS  = 8;      // points per workgroup
constexpr int NE   = PTS * 16; // 128 edges per workgroup

// ---------------------------------------------------------------------------
// Kernel 0: fold the 'c' contraction and the /17 into an effective 64x68
// weight matrix  Wp[o][d] = (1/17) * sum_c W[o][c][d],  zero-padded d=65..67.
// ---------------------------------------------------------------------------
__global__ void wprep_kernel(const float* __restrict__ w, float* __restrict__ wp) {
    int q = blockIdx.x * 256 + threadIdx.x;
    if (q >= COUT * DP) return;
    int o = q / DP, d = q % DP;
    float s = 0.f;
    if (d < DDIM) {
        const float* p = w + (size_t)o * CIN * DDIM + d;
        #pragma unroll 8
        for (int c = 0; c < CIN; ++c) s += p[(size_t)c * DDIM];
        s *= (1.0f / (float)KADJ);
    }
    wp[q] = s;
}

// ---------------------------------------------------------------------------
// Main fused kernel: gather -> frame build -> WMMA GEMM -> atomic scatter.
// One workgroup handles 8 points (128 edges). 8 waves; wave w owns the 16
// edge-columns of point n0+w for the WMMA phase.
// ---------------------------------------------------------------------------
__global__ __launch_bounds__(256) void
spherical_conv_kernel(const float* __restrict__ x, const int* __restrict__ adj,
                      const float* __restrict__ wp, float* __restrict__ out) {
    __shared__ float sF[DP][NE];     // frame matrix (68 x 128), row-major
    __shared__ float sW[COUT][DP];   // effective weight (64 x 68)
    __shared__ float sPart[256];     // sum-of-squares partials
    __shared__ float sS[NE];         // per-edge scale 1/(sqrt2 * r)
    __shared__ float sRS[NE];        // per-edge r*s  (NaN iff r==0, as in ref)
    __shared__ int   sIdx[NE];       // per-edge neighbor index

    const int tid = threadIdx.x;
    const int ntiles = NPTS / PTS;
    const int b  = blockIdx.x / ntiles;
    const int n0 = (blockIdx.x % ntiles) * PTS;

    // Stage effective weight into LDS (4352 floats = 17 * 256).
    #pragma unroll
    for (int i = 0; i < (COUT * DP) / 256; ++i)
        (&sW[0][0])[tid + i * 256] = wp[tid + i * 256];

    // ---- Phase 1: gather + diff + sum of squares (2 threads per edge) ----
    const int e = tid >> 1;          // edge 0..127
    const int h = tid & 1;           // half (channels 0..31 vs 32..63)
    const int p = e >> 4;            // point within tile
    const int j = e & 15;            // neighbor slot (adj column 1+j)
    const int n = n0 + p;
    const int nb = adj[((size_t)b * NPTS + n) * KADJ + 1 + j];
    if (h == 0) sIdx[e] = nb;

    const float* xb = x + (size_t)b * CIN * NPTS;
    float acc = 0.f;
    #pragma unroll
    for (int i = 0; i < 32; ++i) {
        const int d = h * 32 + i;
        const float xc = xb[(size_t)d * NPTS + n];
        const float xn = xb[(size_t)d * NPTS + nb];
        const float df = xn - xc;
        sF[d][e] = df;
        acc += df * df;
    }
    sPart[tid] = acc;
    __syncthreads();

    if (h == 0) {
        const float sum = sPart[tid] + sPart[tid + 1];
        const float r = sqrtf(sum);
        const float s = 1.0f / (1.41421356237309515f * r);  // inf if r==0
        sS[e]  = s;
        sRS[e] = r * s;                                     // 0*inf = NaN (matches ref)
    }
    __syncthreads();

    // ---- Phase 2: normalize frame in LDS, fill row 64 and zero pad rows ----
    #pragma unroll
    for (int i = 0; i < (DP * NE) / 256; ++i) {
        const int q  = tid + i * 256;
        const int d  = q >> 7;        // NE == 128
        const int ee = q & (NE - 1);
        if (d < CIN)       sF[d][ee] *= sS[ee];
        else if (d == CIN) sF[d][ee]  = sRS[ee];
        else               sF[d][ee]  = 0.f;
    }
    __syncthreads();

    // ---- Phase 3: Y(64x16) = Wp(64x68) * F(68x16) per wave, via WMMA f32 ----
    const int wv   = tid >> 5;        // wave id -> column tile
    const int lane = tid & 31;
    const int lh   = lane >> 4;       // lane half selects K = 2*lh, 2*lh+1
    const int lcol = lane & 15;
    const int col  = wv * 16 + lcol;  // edge column owned by this lane

    // Preload all 17 B-fragments (K x 16 tile of F) into registers.
    v2f bf[17];
    #pragma unroll
    for (int kc = 0; kc < 17; ++kc) {
        const int kr = kc * 4 + 2 * lh;
        v2f t;
        t.x = sF[kr][col];
        t.y = sF[kr + 1][col];
        bf[kc] = t;
    }
    const int myidx = sIdx[col];

    #pragma unroll
    for (int ot = 0; ot < 4; ++ot) {
        v8f c = {};
        #pragma unroll
        for (int kc = 0; kc < 17; ++kc) {
            // A fragment: rows ot*16 + (lane%16), K = kc*4 + 2*lh .. +1 (contiguous)
            const v2f a = *(const v2f*)&sW[ot * 16 + lcol][kc * 4 + 2 * lh];
            c = __builtin_amdgcn_wmma_f32_16x16x4_f32(
                    false, a, false, bf[kc], (short)0, c, false, false);
        }
        // D layout: VGPR v -> row ot*16 + v (+8 for lanes 16..31), col = lane%16.
        float* ob = out + ((size_t)b * COUT + ot * 16 + 8 * lh) * NPTS + myidx;
        #pragma unroll
        for (int v = 0; v < 8; ++v)
            unsafeAtomicAdd(ob + (size_t)v * NPTS, c[v]);
    }
}

// ---------------------------------------------------------------------------
extern "C" void kernel_launch(void* const* d_in, const int* in_sizes, int n_in,
                              void* d_out, int out_size, void* d_ws, size_t ws_size,
                              hipStream_t stream) {
    const float* x      = (const float*)d_in[0];
    const int*   adj    = (const int*)d_in[1];
    const float* weight = (const float*)d_in[2];
    float*       out    = (float*)d_out;
    float*       wp     = (float*)d_ws;   // 64*68 floats = 17.4 KB scratch

    // Output is accumulated with atomics: zero it inside the captured stream.
    hipMemsetAsync(d_out, 0, (size_t)out_size * sizeof(float), stream);

    wprep_kernel<<<(COUT * DP + 255) / 256, 256, 0, stream>>>(weight, wp);

    const int nblocks = BB * (NPTS / PTS);  // 16384
    spherical_conv_kernel<<<nblocks, 256, 0, stream>>>(x, adj, wp, out);
}